// NodeAttention_25744033972451
// MI455X (gfx1250) — compile-verified
//
#include <hip/hip_runtime.h>
#include <hip/hip_bf16.h>

typedef __attribute__((ext_vector_type(2))) float     v2f;
typedef __attribute__((ext_vector_type(4))) float     v4f;
typedef __attribute__((ext_vector_type(8))) float     v8f;
typedef __attribute__((ext_vector_type(2))) long long v2ll;

#define FEAT_D   512
#define KCHUNKS  (FEAT_D / 4)   // 128 chunks of K=4 per 16-row tile

// --------------------------------------------------------------------------
// Kernel 1: diag[n] = sigmoid(dot(x[n,:], p) + b) via V_WMMA_F32_16X16X4_F32.
// One wave32 handles a 16-row tile of x. Per K-chunk (4 columns):
//   A (16x4 f32, 2 VGPRs): lane L holds row (tile*16 + (L&15)),
//                          K elements {kk0, kk0+1} with kk0 = (L>>4)*2.
//   B (4x16 f32, 2 VGPRs): p's chunk replicated across all 16 columns ->
//                          every lane just needs p[4*kc + kk0 .. +1].
// C/D (16x16 f32, v8f): every column of D equals the dot product, so
// lane 0 (rows 0..7 in c[0..7]) and lane 16 (rows 8..15) extract results.
// --------------------------------------------------------------------------
__global__ __launch_bounds__(256)
void gate_wmma_kernel(const float* __restrict__ x,
                      const float* __restrict__ p,
                      const float* __restrict__ bias_ptr,
                      float* __restrict__ diag,
                      int N)
{
    __shared__ float p_lds[FEAT_D];
    for (int i = threadIdx.x; i < FEAT_D; i += 256) p_lds[i] = p[i];
    __syncthreads();

    const int lane = threadIdx.x & 31;
    const int wave = threadIdx.x >> 5;
    const int tile = blockIdx.x * 8 + wave;        // 16 rows per tile
    int row = tile * 16 + (lane & 15);
    if (row >= N) row = N - 1;                     // clamp loads; stores guarded
    const int kk0 = (lane >> 4) * 2;               // K offset within chunk

    const float* __restrict__ xr = x + (size_t)row * FEAT_D + kk0;
    const float* __restrict__ pr = p_lds + kk0;

    v8f c = {};
#pragma unroll 4
    for (int kc = 0; kc < KCHUNKS; ++kc) {
        v2f a  = *(const v2f*)(xr + 4 * kc);       // global_load_b64 (clause)
        v2f bb = *(const v2f*)(pr + 4 * kc);       // ds_load_2addr_b64
        // 8 args: (neg_a, A, neg_b, B, c_mod, C, reuse_a, reuse_b)
        c = __builtin_amdgcn_wmma_f32_16x16x4_f32(
                false, a, false, bb, (short)0, c, false, false);
    }

    // Divergence only after all WMMAs (EXEC must be all-ones during WMMA).
    const float bias = bias_ptr[0];
    if ((lane & 15) == 0) {
        const int mbase = tile * 16 + (lane >> 4) * 8;
#pragma unroll
        for (int i = 0; i < 8; ++i) {
            const int r = mbase + i;
            if (r < N) {
                const float t = c[i] + bias;
                // fast reciprocal (v_rcp_f32) instead of IEEE div chain
                diag[r] = __builtin_amdgcn_rcpf(1.0f + __expf(-t));
            }
        }
    }
}

// --------------------------------------------------------------------------
// Kernel 2: streaming edge pass, 4 edges per thread for b128 traffic.
// Coalesced b128 index/attr loads; diag (400 KB) stays L2-resident so the
// random gather is cheap. Also materializes the passthrough edge_index into
// d_out (float output dtype), layout:
//   d_out[0:E)    = (float)edge_index[0][e]
//   d_out[E:2E)   = (float)edge_index[1][e]
//   d_out[2E:3E)  = edge_attr[e] * diag[edge_index[1][e]]
// --------------------------------------------------------------------------
__global__ __launch_bounds__(256)
void edge_scale_kernel(const long long* __restrict__ edge_index,
                       const float* __restrict__ edge_attr,
                       const float* __restrict__ diag,
                       float* __restrict__ out,
                       int E)
{
    const int base = (blockIdx.x * 256 + threadIdx.x) * 4;
    if (base >= E) return;

    if (base + 4 <= E) {
        // fast path: fully vectorized (E % 4 == 0 in practice)
        const v2ll s01 = *(const v2ll*)(edge_index + base);
        const v2ll s23 = *(const v2ll*)(edge_index + base + 2);
        const v2ll c01 = *(const v2ll*)(edge_index + (size_t)E + base);
        const v2ll c23 = *(const v2ll*)(edge_index + (size_t)E + base + 2);
        const v4f  at  = *(const v4f*)(edge_attr + base);

        v4f srcf = { (float)s01.x, (float)s01.y, (float)s23.x, (float)s23.y };
        v4f colf = { (float)c01.x, (float)c01.y, (float)c23.x, (float)c23.y };
        v4f val  = { at.x * diag[(int)c01.x],
                     at.y * diag[(int)c01.y],
                     at.z * diag[(int)c23.x],
                     at.w * diag[(int)c23.y] };

        *(v4f*)(out + base)                 = srcf;
        *(v4f*)(out + (size_t)E + base)     = colf;
        *(v4f*)(out + 2 * (size_t)E + base) = val;
    } else {
        for (int e = base; e < E; ++e) {
            const long long srcn = edge_index[e];
            const long long coln = edge_index[(size_t)E + e];
            const float     attr = edge_attr[e];
            out[e]                 = (float)srcn;
            out[(size_t)E + e]     = (float)coln;
            out[2 * (size_t)E + e] = attr * diag[(int)coln];
        }
    }
}

extern "C" void kernel_launch(void* const* d_in, const int* in_sizes, int n_in,
                              void* d_out, int out_size, void* d_ws, size_t ws_size,
                              hipStream_t stream)
{
    const float*     x          = (const float*)d_in[0];
    const long long* edge_index = (const long long*)d_in[1];
    const float*     edge_attr  = (const float*)d_in[2];
    const float*     p          = (const float*)d_in[3];
    const float*     b          = (const float*)d_in[4];

    const int N = in_sizes[0] / FEAT_D;   // 100000
    const int E = in_sizes[2];            // 6400000

    float* diag = (float*)d_ws;           // N floats of scratch (400 KB)

    const int tiles   = (N + 15) / 16;
    const int blocks1 = (tiles + 7) / 8;  // 8 waves (tiles) per 256-thread block
    gate_wmma_kernel<<<blocks1, 256, 0, stream>>>(x, p, b, diag, N);

    const int blocks2 = (E / 4 + 255) / 256;  // 4 edges per thread
    edge_scale_kernel<<<blocks2, 256, 0, stream>>>(edge_index, edge_attr, diag,
                                                   (float*)d_out, E);
}